// Loss_v2_87729001988706
// MI455X (gfx1250) — compile-verified
//
#include <hip/hip_runtime.h>
#include <math.h>

#define N_ROWS 8192
#define T_COLS 8192
#define BLOCK  256
#define VEC    4
#define CHUNKS (T_COLS / (BLOCK * VEC))   // 8 float4 chunks per thread
#define NWAVES (BLOCK / 32)

__device__ __forceinline__ float warp_max(float v) {
#pragma unroll
    for (int o = 16; o > 0; o >>= 1) v = fmaxf(v, __shfl_xor(v, o, 32));
    return v;
}
__device__ __forceinline__ float warp_sum(float v) {
#pragma unroll
    for (int o = 16; o > 0; o >>= 1) v += __shfl_xor(v, o, 32);
    return v;
}

// CDNA5 async global -> LDS copy (tracked by ASYNCcnt).
__device__ __forceinline__ void async_copy_b128(const float* gsrc, float* ldst) {
    // Generic LDS address low 32 bits == LDS byte offset (aperture mapping).
    unsigned lds_off = (unsigned)(unsigned long long)(void*)ldst;
    asm volatile("global_load_async_to_lds_b128 %0, %1, off"
                 :
                 : "v"(lds_off), "v"(gsrc)
                 : "memory");
}

__device__ __forceinline__ void wait_asynccnt0() {
#if __has_builtin(__builtin_amdgcn_s_wait_asynccnt)
    __builtin_amdgcn_s_wait_asynccnt(0);
#else
    asm volatile("s_wait_asynccnt 0" ::: "memory");
#endif
}

__global__ __launch_bounds__(BLOCK) void row_loss_kernel(
    const float* __restrict__ input, const float* __restrict__ label,
    float* __restrict__ row_out)
{
    __shared__ float rowbuf[T_COLS];     // 32 KB: one full row
    __shared__ float redmax[NWAVES];
    __shared__ float redsum[NWAVES];
    __shared__ float bcast_max;

    const int row  = blockIdx.x;
    const int tid  = threadIdx.x;
    const int lane = tid & 31;
    const int wid  = tid >> 5;

    const float* grow = input + (size_t)row * T_COLS;

    // ---- Stage row into LDS via the async copy engine (ASYNCcnt path) ----
#pragma unroll
    for (int k = 0; k < CHUNKS; ++k) {
        const int fidx = (tid + k * BLOCK) * VEC;   // coalesced 16B/lane
        async_copy_b128(grow + fidx, &rowbuf[fidx]);
    }
    wait_asynccnt0();     // this wave's copies done
    __syncthreads();      // all waves' copies done -> LDS row valid

    const float4* rowv = reinterpret_cast<const float4*>(rowbuf);

    // ---- Pass 1: row max (from LDS) ----
    float m = -INFINITY;
#pragma unroll
    for (int k = 0; k < CHUNKS; ++k) {
        float4 v = rowv[tid + k * BLOCK];
        m = fmaxf(m, fmaxf(fmaxf(v.x, v.y), fmaxf(v.z, v.w)));
    }
    m = warp_max(m);
    if (lane == 0) redmax[wid] = m;
    __syncthreads();
    if (wid == 0) {
        float x = (lane < NWAVES) ? redmax[lane] : -INFINITY;
        x = warp_max(x);
        if (lane == 0) bcast_max = x;
    }
    __syncthreads();
    const float rowmax = bcast_max;

    // ---- Pass 2: sum of exp(x - max) (from LDS) ----
    float s = 0.0f;
#pragma unroll
    for (int k = 0; k < CHUNKS; ++k) {
        float4 v = rowv[tid + k * BLOCK];
        s += __expf(v.x - rowmax) + __expf(v.y - rowmax)
           + __expf(v.z - rowmax) + __expf(v.w - rowmax);
    }
    s = warp_sum(s);
    if (lane == 0) redsum[wid] = s;
    __syncthreads();

    if (tid == 0) {
        float tot = 0.0f;
#pragma unroll
        for (int i = 0; i < NWAVES; ++i) tot += redsum[i];

        // ---- Sparse target with the reference's overwrite semantics ----
        // writes in order: (c1,0.1) (c2,wB) (c3,0.1) (c4,wD);
        // a write survives only if no LATER write hits the same column.
        const float lab = label[row];
        const float pos = lab * (float)T_COLS - 1.0f;
        const int fl = (int)floorf(pos);
        const int ce = (int)ceilf(pos);
        const int c1 = max(fl - 1, 0);
        const int c2 = fl;
        const int c3 = min(ce + 1, T_COLS - 1);
        const int c4 = ce;
        const float w1 = 0.1f;
        const float w2 = (fl >= 1) ? 0.4f : 0.5f;
        const float w3 = 0.1f;
        const float w4 = (ce < T_COLS - 1) ? 0.4f : 0.5f;
        const bool l1 = (c1 != c2) && (c1 != c3) && (c1 != c4);
        const bool l2 = (c2 != c3) && (c2 != c4);
        const bool l3 = (c3 != c4);

        float wsum = w4, dot = w4 * rowbuf[c4];
        if (l1) { wsum += w1; dot += w1 * rowbuf[c1]; }
        if (l2) { wsum += w2; dot += w2 * rowbuf[c2]; }
        if (l3) { wsum += w3; dot += w3 * rowbuf[c3]; }

        // loss_row = wsum*(max + log(sum exp(x-max))) - sum(w*x)
        row_out[row] = wsum * (rowmax + __logf(tot)) - dot;
    }
}

__global__ __launch_bounds__(BLOCK) void mean_kernel(
    const float* __restrict__ rows, float* __restrict__ out)
{
    __shared__ float red[NWAVES];
    float s = 0.0f;
    for (int i = threadIdx.x; i < N_ROWS; i += BLOCK) s += rows[i];
    s = warp_sum(s);
    const int lane = threadIdx.x & 31, wid = threadIdx.x >> 5;
    if (lane == 0) red[wid] = s;
    __syncthreads();
    if (threadIdx.x == 0) {
        float tot = 0.0f;
#pragma unroll
        for (int i = 0; i < NWAVES; ++i) tot += red[i];
        out[0] = tot / (float)N_ROWS;
    }
}

extern "C" void kernel_launch(void* const* d_in, const int* in_sizes, int n_in,
                              void* d_out, int out_size, void* d_ws, size_t ws_size,
                              hipStream_t stream) {
    const float* input = (const float*)d_in[0];   // [8192, 8192] fp32
    const float* label = (const float*)d_in[1];   // [8192] fp32
    float* out = (float*)d_out;                   // scalar fp32
    float* ws  = (float*)d_ws;                    // >= 8192 floats

    row_loss_kernel<<<N_ROWS, BLOCK, 0, stream>>>(input, label, ws);
    mean_kernel<<<1, BLOCK, 0, stream>>>(ws, out);
}